// SimpleGNN_79388175499762
// MI455X (gfx1250) — compile-verified
//
#include <hip/hip_runtime.h>
#include <hip/hip_bf16.h>

#define N_NODES 8192
#define N_EDGES 262144
#define S_DIM 5
#define F_DIM 7
#define ROUNDS 4

typedef __attribute__((ext_vector_type(2))) float v2f;
typedef __attribute__((ext_vector_type(8))) float v8f;

// sigmoid(v) = 1/(1+e^-v) using hardware v_rcp_f32 (~1 ulp) instead of the
// IEEE div_scale/Newton chain the compiler emits for '/'.
__device__ __forceinline__ float sigmoidf_fast(float v) {
    return __builtin_amdgcn_rcpf(1.0f + __expf(-v));
}

// ---------------------------------------------------------------------------
// 1) dropout + input MLP: state = relu(xd @ in_W + in_b)
// ---------------------------------------------------------------------------
__global__ void gnn_init_kernel(const float* __restrict__ x,
                                const float* __restrict__ drop_u,
                                const float* __restrict__ in_W,  // [7,5]
                                const float* __restrict__ in_b,  // [5]
                                float* __restrict__ state) {
    int i = blockIdx.x * blockDim.x + threadIdx.x;
    if (i >= N_NODES) return;
    float xd[F_DIM];
#pragma unroll
    for (int j = 0; j < F_DIM; ++j) {
        float v = x[i * F_DIM + j];
        xd[j] = (drop_u[i * F_DIM + j] > 0.25f) ? v * (1.0f / 0.75f) : 0.0f;
    }
#pragma unroll
    for (int s = 0; s < S_DIM; ++s) {
        float acc = in_b[s];
#pragma unroll
        for (int j = 0; j < F_DIM; ++j) acc = fmaf(xd[j], in_W[j * S_DIM + s], acc);
        state[i * S_DIM + s] = fmaxf(acc, 0.0f);
    }
}

// ---------------------------------------------------------------------------
// 2) message = relu(state @ msg_W + msg_b); also zero agg for this round
// ---------------------------------------------------------------------------
__global__ void gnn_message_kernel(const float* __restrict__ state,
                                   const float* __restrict__ msg_W,  // [5,5]
                                   const float* __restrict__ msg_b,  // [5]
                                   float* __restrict__ message,
                                   float* __restrict__ agg) {
    int i = blockIdx.x * blockDim.x + threadIdx.x;
    if (i >= N_NODES) return;
    float st[S_DIM];
#pragma unroll
    for (int k = 0; k < S_DIM; ++k) st[k] = state[i * S_DIM + k];
#pragma unroll
    for (int s = 0; s < S_DIM; ++s) {
        float acc = msg_b[s];
#pragma unroll
        for (int k = 0; k < S_DIM; ++k) acc = fmaf(st[k], msg_W[k * S_DIM + s], acc);
        message[i * S_DIM + s] = fmaxf(acc, 0.0f);
        agg[i * S_DIM + s] = 0.0f;
    }
}

// ---------------------------------------------------------------------------
// 3) scatter-add: agg[dst] += message[src]
// ---------------------------------------------------------------------------
__global__ void gnn_scatter_kernel(const int* __restrict__ src,
                                   const int* __restrict__ dst,
                                   const float* __restrict__ message,
                                   float* __restrict__ agg) {
    int e = blockIdx.x * blockDim.x + threadIdx.x;
    if (e >= N_EDGES) return;
    int s = src[e];
    int d = dst[e];
#pragma unroll
    for (int j = 0; j < S_DIM; ++j) {
        atomicAdd(&agg[d * S_DIM + j], message[s * S_DIM + j]);
    }
}

// ---------------------------------------------------------------------------
// 4) GRU cell update
// ---------------------------------------------------------------------------
__global__ void gnn_gru_kernel(float* __restrict__ state,
                               const float* __restrict__ agg,
                               const float* __restrict__ rW, const float* __restrict__ rb,
                               const float* __restrict__ uW, const float* __restrict__ ub,
                               const float* __restrict__ tW, const float* __restrict__ tb) {
    int i = blockIdx.x * blockDim.x + threadIdx.x;
    if (i >= N_NODES) return;
    float st[S_DIM], ag[S_DIM];
#pragma unroll
    for (int k = 0; k < S_DIM; ++k) {
        st[k] = state[i * S_DIM + k];
        ag[k] = agg[i * S_DIM + k];
    }
    float reset[S_DIM], update[S_DIM];
#pragma unroll
    for (int s = 0; s < S_DIM; ++s) {
        float r = rb[s];
        float u = ub[s];
#pragma unroll
        for (int k = 0; k < S_DIM; ++k) {
            r = fmaf(st[k], rW[k * S_DIM + s], r);
            u = fmaf(st[k], uW[k * S_DIM + s], u);
        }
#pragma unroll
        for (int k = 0; k < S_DIM; ++k) {
            r = fmaf(ag[k], rW[(S_DIM + k) * S_DIM + s], r);
            u = fmaf(ag[k], uW[(S_DIM + k) * S_DIM + s], u);
        }
        reset[s] = sigmoidf_fast(r);
        update[s] = sigmoidf_fast(u);
    }
#pragma unroll
    for (int s = 0; s < S_DIM; ++s) {
        float c = tb[s];
#pragma unroll
        for (int k = 0; k < S_DIM; ++k) c = fmaf(st[k] * reset[k], tW[k * S_DIM + s], c);
#pragma unroll
        for (int k = 0; k < S_DIM; ++k) c = fmaf(ag[k], tW[(S_DIM + k) * S_DIM + s], c);
        float cand = tanhf(c);
        state[i * S_DIM + s] = st[s] * (1.0f - update[s]) + cand * update[s];
    }
}

// ---------------------------------------------------------------------------
// 5) head: mean / logvar / z = mean + eps * exp(0.5*logvar)
// ---------------------------------------------------------------------------
__global__ void gnn_head_kernel(const float* __restrict__ state,
                                const float* __restrict__ eps,
                                const float* __restrict__ muW, const float* __restrict__ mub,
                                const float* __restrict__ lvW, const float* __restrict__ lvb,
                                float* __restrict__ z,
                                float* __restrict__ mean,
                                float* __restrict__ logvar) {
    int i = blockIdx.x * blockDim.x + threadIdx.x;
    if (i >= N_NODES) return;
    float st[S_DIM];
#pragma unroll
    for (int k = 0; k < S_DIM; ++k) st[k] = state[i * S_DIM + k];
#pragma unroll
    for (int s = 0; s < S_DIM; ++s) {
        float m = mub[s];
        float lv = lvb[s];
#pragma unroll
        for (int k = 0; k < S_DIM; ++k) {
            m = fmaf(st[k], muW[k * S_DIM + s], m);
            lv = fmaf(st[k], lvW[k * S_DIM + s], lv);
        }
        float zz = fmaf(eps[i * S_DIM + s], __expf(0.5f * lv), m);
        mean[i * S_DIM + s] = m;
        logvar[i * S_DIM + s] = lv;
        z[i * S_DIM + s] = zz;
    }
}

// ---------------------------------------------------------------------------
// 6) dense decoder: z_dc = sigmoid(z @ z^T), [8192 x 8192]
//    One wave per 16x16 output tile; K=5 done as WMMA(K=0..3) + padded
//    WMMA with K=4 in slot 0.  V_WMMA_F32_16X16X4_F32 operand layouts:
//      A 16x4:  v0 = A[m][0] (lanes 0-15) / A[m][2] (lanes 16-31)
//               v1 = A[m][1]              / A[m][3]
//      B 4x16:  v0 = B[0][n] (lanes 0-15) / B[2][n] (lanes 16-31)
//               v1 = B[1][n]              / B[3][n]     (B[k][n] = z[col+n][k])
//      C/D:     VGPR v: row = v + 8*(lane>=16), col = lane&15
// ---------------------------------------------------------------------------
__global__ void __launch_bounds__(256) zdc_wmma_kernel(const float* __restrict__ z,
                                                       float* __restrict__ zdc) {
    const int lane = threadIdx.x & 31;
    const int wave = threadIdx.x >> 5;
    const int tile = blockIdx.x * 8 + wave;          // 0 .. 512*512-1
    const int tileRow = tile >> 9;                   // tile / 512
    const int tileCol = tile & 511;                  // tile % 512
    const int rowBase = tileRow * 16;
    const int colBase = tileCol * 16;
    const int m = lane & 15;
    const bool hi = lane >= 16;
    const int kA = hi ? 2 : 0;

    // A operand: rows of z (this tile's 16 rows), K slots 0..3
    v2f a;
    a.x = z[(rowBase + m) * S_DIM + kA + 0];
    a.y = z[(rowBase + m) * S_DIM + kA + 1];
    // B operand: z^T -> B[k][n] = z[colBase+n][k]
    v2f b;
    b.x = z[(colBase + m) * S_DIM + kA + 0];
    b.y = z[(colBase + m) * S_DIM + kA + 1];

    v8f acc = {};
    acc = __builtin_amdgcn_wmma_f32_16x16x4_f32(
        /*neg_a=*/false, a, /*neg_b=*/false, b,
        /*c_mod=*/(short)0, acc, /*reuse_a=*/false, /*reuse_b=*/false);

    // K=4 fixup: pad a second 16x4 x 4x16 product with K=4 in slot 0, zeros elsewhere
    float a4 = z[(rowBase + m) * S_DIM + 4];
    float b4 = z[(colBase + m) * S_DIM + 4];
    v2f a2, b2;
    a2.x = hi ? 0.0f : a4;
    a2.y = 0.0f;
    b2.x = hi ? 0.0f : b4;
    b2.y = 0.0f;
    acc = __builtin_amdgcn_wmma_f32_16x16x4_f32(
        false, a2, false, b2, (short)0, acc, false, false);

    // sigmoid + store (row = rowBase + v + 8*hi, col = colBase + (lane&15))
    const int col = colBase + m;
    const int rOff = hi ? 8 : 0;
#pragma unroll
    for (int v = 0; v < 8; ++v) {
        int row = rowBase + v + rOff;
        float val = acc[v];
        zdc[(size_t)row * N_NODES + col] = sigmoidf_fast(val);
    }
}

// ---------------------------------------------------------------------------
// launch
// ---------------------------------------------------------------------------
extern "C" void kernel_launch(void* const* d_in, const int* in_sizes, int n_in,
                              void* d_out, int out_size, void* d_ws, size_t ws_size,
                              hipStream_t stream) {
    const float* x      = (const float*)d_in[0];
    const float* drop_u = (const float*)d_in[1];
    const float* eps    = (const float*)d_in[2];
    const int*   eidx   = (const int*)d_in[3];
    // d_in[4] = batch (unused by reference)
    const float* in_W   = (const float*)d_in[5];
    const float* in_b   = (const float*)d_in[6];
    const float* msg_W  = (const float*)d_in[7];   // [4,5,5]
    const float* msg_b  = (const float*)d_in[8];   // [4,5]
    const float* gru_rW = (const float*)d_in[9];   // [4,10,5]
    const float* gru_rb = (const float*)d_in[10];
    const float* gru_uW = (const float*)d_in[11];
    const float* gru_ub = (const float*)d_in[12];
    const float* gru_tW = (const float*)d_in[13];
    const float* gru_tb = (const float*)d_in[14];
    const float* mu_W   = (const float*)d_in[15];
    const float* mu_b   = (const float*)d_in[16];
    const float* lv_W   = (const float*)d_in[17];
    const float* lv_b   = (const float*)d_in[18];

    // d_out layout (return order): z [N,5], z_dc [N,N], mean [N,5], logvar [N,5]
    float* out    = (float*)d_out;
    float* z      = out;
    float* zdc    = out + (size_t)N_NODES * S_DIM;
    float* mean   = out + (size_t)N_NODES * S_DIM + (size_t)N_NODES * N_NODES;
    float* logvar = mean + (size_t)N_NODES * S_DIM;

    float* ws      = (float*)d_ws;
    float* state   = ws;                       // [N,5]
    float* message = ws + N_NODES * S_DIM;     // [N,5]
    float* agg     = ws + 2 * N_NODES * S_DIM; // [N,5]

    const int NTHD = 128;
    const int NBLK = (N_NODES + NTHD - 1) / NTHD;

    gnn_init_kernel<<<NBLK, NTHD, 0, stream>>>(x, drop_u, in_W, in_b, state);

    const int* src = eidx;
    const int* dst = eidx + N_EDGES;
    for (int r = 0; r < ROUNDS; ++r) {
        gnn_message_kernel<<<NBLK, NTHD, 0, stream>>>(
            state, msg_W + r * S_DIM * S_DIM, msg_b + r * S_DIM, message, agg);
        gnn_scatter_kernel<<<(N_EDGES + 255) / 256, 256, 0, stream>>>(
            src, dst, message, agg);
        gnn_gru_kernel<<<NBLK, NTHD, 0, stream>>>(
            state, agg,
            gru_rW + r * 2 * S_DIM * S_DIM, gru_rb + r * S_DIM,
            gru_uW + r * 2 * S_DIM * S_DIM, gru_ub + r * S_DIM,
            gru_tW + r * 2 * S_DIM * S_DIM, gru_tb + r * S_DIM);
    }

    gnn_head_kernel<<<NBLK, NTHD, 0, stream>>>(
        state, eps, mu_W, mu_b, lv_W, lv_b, z, mean, logvar);

    // 512x512 tiles of 16x16, 8 waves (tiles) per 256-thread block
    zdc_wmma_kernel<<<(512 * 512) / 8, 256, 0, stream>>>(z, zdc);
}